// LocalGrouper_21397527069059
// MI455X (gfx1250) — compile-verified
//
#include <hip/hip_runtime.h>
#include <hip/hip_bf16.h>

#define N_PTS 8192
#define S_PTS 2048
#define KNN_K 32

typedef __attribute__((ext_vector_type(2))) float v2f;
typedef __attribute__((ext_vector_type(8))) float v8f;
typedef unsigned int u32;
typedef __attribute__((ext_vector_type(4))) unsigned int v4u;
typedef __attribute__((ext_vector_type(8))) unsigned int v8u;

// ---------------------------------------------------------------------------
// Kernel 1: Farthest point sampling. One workgroup per batch; running min
// distance array lives in LDS (32 KB). 2048 dependent steps of
// distance-update + block argmax.
// ---------------------------------------------------------------------------
__global__ __launch_bounds__(256) void fps_kernel(const float* __restrict__ xyz,
                                                  int* __restrict__ fps_idx) {
    __shared__ float dist[N_PTS];
    __shared__ float rd[256];
    __shared__ int   ri[256];
    __shared__ int   cur;
    const int b   = blockIdx.x;
    const int tid = threadIdx.x;
    const float* base = xyz + (size_t)b * N_PTS * 3;

    for (int j = tid; j < N_PTS; j += 256) dist[j] = 1e10f;
    if (tid == 0) cur = 0;
    __syncthreads();

    for (int it = 0; it < S_PTS; ++it) {
        const int c = cur;
        if (tid == 0) fps_idx[b * S_PTS + it] = c;
        const float cx = base[c * 3 + 0];
        const float cy = base[c * 3 + 1];
        const float cz = base[c * 3 + 2];
        float bestv = -1.0f;
        int   besti = 0;
        for (int j = tid; j < N_PTS; j += 256) {
            const float dx = base[j * 3 + 0] - cx;
            const float dy = base[j * 3 + 1] - cy;
            const float dz = base[j * 3 + 2] - cz;
            const float d  = dx * dx + dy * dy + dz * dz;
            const float m  = fminf(dist[j], d);
            dist[j] = m;
            if (m > bestv) { bestv = m; besti = j; }
        }
        rd[tid] = bestv;
        ri[tid] = besti;
        __syncthreads();
        for (int sft = 128; sft > 0; sft >>= 1) {
            if (tid < sft) {
                const float ov = rd[tid + sft];
                const int   oi = ri[tid + sft];
                if (ov > rd[tid] || (ov == rd[tid] && oi < ri[tid])) {
                    rd[tid] = ov; ri[tid] = oi;
                }
            }
            __syncthreads();
        }
        if (tid == 0) cur = ri[0];
        __syncthreads();
    }
}

// ---------------------------------------------------------------------------
// Kernel 2: gather sampled xyz/feat into the output buffer.
// ---------------------------------------------------------------------------
__global__ __launch_bounds__(64) void gather_sampled(const float* __restrict__ xyz,
                                                     const float* __restrict__ feat,
                                                     const int* __restrict__ fps_idx,
                                                     float* __restrict__ out_xyz_s,
                                                     float* __restrict__ out_feat_s) {
    const int gid = blockIdx.x;            // b * S_PTS + s
    const int b   = gid / S_PTS;
    const int t   = threadIdx.x;
    const int src = fps_idx[gid];
    const float* fsrc = feat + ((size_t)b * N_PTS + src) * 64;
    float*       fdst = out_feat_s + (size_t)gid * 64;
    fdst[t] = fsrc[t];
    if (t < 3) out_xyz_s[(size_t)gid * 3 + t] =
        xyz[((size_t)b * N_PTS + src) * 3 + t];
}

// ---------------------------------------------------------------------------
// Kernel 3: KNN. One wave per 16-query tile.
//  - B-tile (16 points x 3 f32) is streamed Global->LDS by the Tensor Data
//    Mover (tensor_load_to_lds + s_wait_tensorcnt), descriptor per ISA §8.
//  - q.x dot products via V_WMMA_F32_16X16X4_F32 (K padded 3->4 with zeros).
//  - per-query top-32 lists kept in LDS (dynamic indexing -> ds ops, not
//    cndmask select trees).
// ---------------------------------------------------------------------------
__global__ __launch_bounds__(32) void knn_kernel(const float* __restrict__ xyz,
                                                 const float* __restrict__ xyz_s,
                                                 int* __restrict__ idx_knn) {
    const int lane = threadIdx.x;
    const int qt   = blockIdx.x & 127;   // query tile within batch (2048/16)
    const int b    = blockIdx.x >> 7;
    const float* pts = xyz   + (size_t)b * N_PTS * 3;
    const float* qs  = xyz_s + ((size_t)b * S_PTS + qt * 16) * 3;

    __shared__ float q2s[16];
    __shared__ float dt[16][17];
    __shared__ float ptile[48];          // TDM destination: 16 pts x 3 coords
    __shared__ float bd[16][KNN_K];      // per-query top-k distances
    __shared__ int   bi[16][KNN_K];      // per-query top-k indices

    // A-matrix (16x4 f32): lane L holds row M=L%16, K pair (0,1) for lanes
    // 0-15 and (2,3) for lanes 16-31, one K per VGPR.
    const int m  = lane & 15;
    const int kh = (lane >> 4) * 2;
    const float qc[4] = { qs[m * 3 + 0], qs[m * 3 + 1], qs[m * 3 + 2], 0.0f };
    v2f A;
    A.x = qc[kh];
    A.y = qc[kh + 1];
    if (lane < 16) {
        q2s[m] = qc[0] * qc[0] + qc[1] * qc[1] + qc[2] * qc[2];
        for (int i = 0; i < KNN_K; ++i) { bd[m][i] = 1e30f; bi[m][i] = 0; }
    }
    __syncthreads();

    // --- TDM descriptor (D#), CDNA5 ISA ---
    // Group 1 is loop-invariant: data_size=4B, 2D tensor [8192 rows x 3],
    // tile = 3 x 16 rows, dim0 stride = 3 elements.
    const u32 ldsoff = (u32)(size_t)&ptile[0];
    v8u g1;
    g1[0] = 0x2u << 16;                 // workgroup_mask=0, data_size=2 (4B)
    g1[1] = 3u << 16;                   // tensor_dim0 = 3 (low 16 bits)
    g1[2] = (N_PTS & 0xFFFFu) << 16;    // tensor_dim0 hi=0; tensor_dim1 lo=8192
    g1[3] = 3u << 16;                   // tensor_dim1 hi=0; tile_dim0 = 3
    g1[4] = 16u;                        // tile_dim1 = 16, tile_dim2 = 0
    g1[5] = 3u;                         // tensor_dim0_stride = 3 (low 32)
    g1[6] = 0u;                         // stride0 hi=0; dim1_stride lo=0
    g1[7] = 0u;

    float worst     = 1e30f;
    int   worst_pos = 0;
    const int mhalf = (lane >> 4) * 8;
    const int n     = lane & 15;

    for (int nt = 0; nt < N_PTS / 16; ++nt) {
        // Group 0: count=1, lds_addr, global_addr of tile start, type=2.
        const unsigned long long gaddr =
            (unsigned long long)(size_t)pts + (unsigned long long)nt * 192ull;
        v4u g0;
        g0[0] = 0x1u;                                   // count=1, user mode
        g0[1] = ldsoff;                                 // lds_addr (bytes)
        g0[2] = (u32)(gaddr & 0xFFFFFFFFu);             // global_addr[31:0]
        g0[3] = (u32)((gaddr >> 32) & 0x01FFFFFFu)      // global_addr[56:32]
              | (2u << 30);                             // type=2 (image)

        asm volatile("tensor_load_to_lds %0, %1" :: "s"(g0), "s"(g1) : "memory");

        if (nt + 1 < N_PTS / 16)
            __builtin_prefetch(pts + (size_t)(nt + 1) * 48, 0, 1);

        __builtin_amdgcn_s_wait_tensorcnt(0);

        // B-matrix (4x16 f32) from the LDS tile; same K pairing as A.
        const float px = ptile[n * 3 + 0];
        const float py = ptile[n * 3 + 1];
        const float pz = ptile[n * 3 + 2];
        const float pc[4] = { px, py, pz, 0.0f };
        v2f B;
        B.x = pc[kh];
        B.y = pc[kh + 1];
        const float x2 = px * px + py * py + pz * pz;

        v8f C = {};
        v8f D = __builtin_amdgcn_wmma_f32_16x16x4_f32(
            /*neg_a=*/false, A, /*neg_b=*/false, B,
            /*c_mod=*/(short)0, C, /*reuse_a=*/false, /*reuse_b=*/false);

#pragma unroll
        for (int r = 0; r < 8; ++r) {
            dt[mhalf + r][n] = q2s[mhalf + r] + x2 - 2.0f * D[r];
        }
        __syncthreads();

        if (lane < 16) {
            for (int j = 0; j < 16; ++j) {
                const float dd = dt[lane][j];
                if (dd < worst) {
                    bd[lane][worst_pos] = dd;          // LDS-resident list:
                    bi[lane][worst_pos] = nt * 16 + j; // dynamic index -> ds op
                    worst = bd[lane][0];
                    worst_pos = 0;
                    for (int t2 = 1; t2 < KNN_K; ++t2) {
                        const float v = bd[lane][t2];
                        if (v > worst) { worst = v; worst_pos = t2; }
                    }
                }
            }
        }
        __syncthreads();
    }

    if (lane < 16) {
        // ascending insertion sort in LDS (matches top_k smallest-first)
        for (int i = 1; i < KNN_K; ++i) {
            const float dv = bd[lane][i];
            const int   iv = bi[lane][i];
            int j = i - 1;
            while (j >= 0 && bd[lane][j] > dv) {
                bd[lane][j + 1] = bd[lane][j];
                bi[lane][j + 1] = bi[lane][j];
                --j;
            }
            bd[lane][j + 1] = dv;
            bi[lane][j + 1] = iv;
        }
        int* dst = idx_knn + (((size_t)b * S_PTS) + qt * 16 + lane) * KNN_K;
        for (int i = 0; i < KNN_K; ++i) dst[i] = bi[lane][i];
    }
}

// ---------------------------------------------------------------------------
// Stats / normalization kernels: per-k-slot sum & sumsq of diffs across
// (b, s, c), hierarchical LDS partials -> global atomics.
// ---------------------------------------------------------------------------
__global__ __launch_bounds__(192) void zero_stats(float* __restrict__ stats) {
    const int t = threadIdx.x;
    if (t < 128) stats[t] = 0.0f;
}

__global__ __launch_bounds__(96) void stats_xyz(const float* __restrict__ xyz,
                                                const float* __restrict__ xyz_s,
                                                const int* __restrict__ idx_knn,
                                                float* __restrict__ sum,
                                                float* __restrict__ sumsq) {
    const int gid = blockIdx.x;
    const int b   = gid / S_PTS;
    const int t   = threadIdx.x;      // 96 = 32k * 3c
    const int k   = t / 3;
    const int c   = t % 3;
    __shared__ float ls[32], lq[32];
    if (t < 32) { ls[t] = 0.0f; lq[t] = 0.0f; }
    __syncthreads();
    const int nidx = idx_knn[(size_t)gid * KNN_K + k];
    const float v  = xyz[((size_t)b * N_PTS + nidx) * 3 + c]
                   - xyz_s[(size_t)gid * 3 + c];
    atomicAdd(&ls[k], v);
    atomicAdd(&lq[k], v * v);
    __syncthreads();
    if (t < 32) { atomicAdd(&sum[t], ls[t]); atomicAdd(&sumsq[t], lq[t]); }
}

__global__ __launch_bounds__(256) void stats_feat(const float* __restrict__ feat,
                                                  const float* __restrict__ feat_s,
                                                  const int* __restrict__ idx_knn,
                                                  float* __restrict__ sum,
                                                  float* __restrict__ sumsq) {
    const int gid = blockIdx.x;
    const int b   = gid / S_PTS;
    const int t   = threadIdx.x;
    __shared__ float ls[32], lq[32];
    if (t < 32) { ls[t] = 0.0f; lq[t] = 0.0f; }
    __syncthreads();
    for (int e = t; e < 32 * 64; e += 256) {
        const int k = e >> 6;
        const int c = e & 63;
        const int nidx = idx_knn[(size_t)gid * KNN_K + k];
        const float v  = feat[((size_t)b * N_PTS + nidx) * 64 + c]
                       - feat_s[(size_t)gid * 64 + c];
        atomicAdd(&ls[k], v);
        atomicAdd(&lq[k], v * v);
    }
    __syncthreads();
    if (t < 32) { atomicAdd(&sum[t], ls[t]); atomicAdd(&sumsq[t], lq[t]); }
}

// stats layout: [sum_xyz 0:32][sumsq_xyz 32:64][sum_feat 64:96][sumsq_feat 96:128]
// scales layout: [xyz 0:32][feat 32:64]
__global__ __launch_bounds__(64) void compute_scales(const float* __restrict__ stats,
                                                     float* __restrict__ scales) {
    const int t = threadIdx.x;
    const int k = t & 31;
    const bool isfeat = t >= 32;
    const float N  = isfeat ? (float)(8 * S_PTS * 64) : (float)(8 * S_PTS * 3);
    const float s  = stats[(isfeat ? 64 : 0) + k];
    const float sq = stats[(isfeat ? 96 : 32) + k];
    float var = (sq - s * s / N) / (N - 1.0f);   // unbiased (ddof=1)
    var = fmaxf(var, 0.0f);
    float sd = sqrtf(var);
    sd = fmaxf(sd, 1e-5f);                        // clip(std, 1e-5)
    scales[t] = 1.0f / sd;
}

// ---------------------------------------------------------------------------
// Output kernels (bandwidth bound)
// ---------------------------------------------------------------------------
__global__ __launch_bounds__(96) void write_xyz_knn(const float* __restrict__ xyz,
                                                    const float* __restrict__ xyz_s,
                                                    const int* __restrict__ idx_knn,
                                                    const float* __restrict__ scales,
                                                    float* __restrict__ out) {
    const int gid = blockIdx.x;
    const int b   = gid / S_PTS;
    const int t   = threadIdx.x;
    const int k   = t / 3;
    const int c   = t % 3;
    const int nidx = idx_knn[(size_t)gid * KNN_K + k];
    const float v  = xyz[((size_t)b * N_PTS + nidx) * 3 + c]
                   - xyz_s[(size_t)gid * 3 + c];
    out[((size_t)gid * KNN_K + k) * 3 + c] = v * scales[k];
}

__global__ __launch_bounds__(256) void write_feat_knn(const float* __restrict__ feat,
                                                      const float* __restrict__ feat_s,
                                                      const int* __restrict__ idx_knn,
                                                      const float* __restrict__ scales,
                                                      float* __restrict__ out) {
    const int gid = blockIdx.x;
    const int b   = gid / S_PTS;
    const int t   = threadIdx.x;
    for (int e = t; e < 32 * 128; e += 256) {
        const int k = e >> 7;
        const int c = e & 127;
        float v;
        if (c < 64) {
            const int nidx = idx_knn[(size_t)gid * KNN_K + k];
            v = (feat[((size_t)b * N_PTS + nidx) * 64 + c]
                 - feat_s[(size_t)gid * 64 + c]) * scales[32 + k];
        } else {
            v = feat_s[(size_t)gid * 64 + (c - 64)];   // center replicated
        }
        out[((size_t)gid * KNN_K + k) * 128 + c] = v;
    }
}

// ---------------------------------------------------------------------------
// Launch
// ---------------------------------------------------------------------------
extern "C" void kernel_launch(void* const* d_in, const int* in_sizes, int n_in,
                              void* d_out, int out_size, void* d_ws, size_t ws_size,
                              hipStream_t stream) {
    const float* xyz  = (const float*)d_in[0];   // [8, 8192, 3]
    const float* feat = (const float*)d_in[1];   // [8, 8192, 64]

    const int B = 8;

    // output layout (floats, concatenated in return order)
    float* out          = (float*)d_out;
    float* out_xyz_s    = out;                                   // [8,2048,3]
    float* out_feat_s   = out + (size_t)B * S_PTS * 3;           // [8,2048,64]
    float* out_xyz_knn  = out_feat_s + (size_t)B * S_PTS * 64;   // [8,2048,32,3]
    float* out_feat_knn = out_xyz_knn + (size_t)B * S_PTS * 32 * 3; // [8,2048,32,128]

    // workspace layout
    int*   fps_idx = (int*)d_ws;                                 // [8,2048]
    int*   idx_knn = fps_idx + (size_t)B * S_PTS;                // [8,2048,32]
    float* stats   = (float*)(idx_knn + (size_t)B * S_PTS * KNN_K); // 128 floats
    float* scales  = stats + 128;                                // 64 floats

    float* sum_xyz    = stats;
    float* sumsq_xyz  = stats + 32;
    float* sum_feat   = stats + 64;
    float* sumsq_feat = stats + 96;

    fps_kernel<<<B, 256, 0, stream>>>(xyz, fps_idx);
    gather_sampled<<<B * S_PTS, 64, 0, stream>>>(xyz, feat, fps_idx,
                                                 out_xyz_s, out_feat_s);
    knn_kernel<<<B * (S_PTS / 16), 32, 0, stream>>>(xyz, out_xyz_s, idx_knn);
    zero_stats<<<1, 192, 0, stream>>>(stats);
    stats_xyz<<<B * S_PTS, 96, 0, stream>>>(xyz, out_xyz_s, idx_knn,
                                            sum_xyz, sumsq_xyz);
    stats_feat<<<B * S_PTS, 256, 0, stream>>>(feat, out_feat_s, idx_knn,
                                              sum_feat, sumsq_feat);
    compute_scales<<<1, 64, 0, stream>>>(stats, scales);
    write_xyz_knn<<<B * S_PTS, 96, 0, stream>>>(xyz, out_xyz_s, idx_knn,
                                                scales, out_xyz_knn);
    write_feat_knn<<<B * S_PTS, 256, 0, stream>>>(feat, out_feat_s, idx_knn,
                                                  scales, out_feat_knn);
}